// MD_MMD_Layer_84232898609624
// MI455X (gfx1250) — compile-verified
//
#include <hip/hip_runtime.h>
#include <math.h>

typedef __attribute__((ext_vector_type(2))) float v2f;
typedef __attribute__((ext_vector_type(8))) float v8f;

#define B_SZ   4096
#define IN_SZ  512
#define OUT_SZ 256
#define NTOT   8192          // 2*B
#define NT64   128           // 8192/64 tiles per dim in pair pass

static __device__ __forceinline__ v8f wmma_f32(v2f a, v2f b, v8f c) {
    return __builtin_amdgcn_wmma_f32_16x16x4_f32(false, a, false, b, (short)0, c, false, false);
}

// ---------------- init scratch (colsum + reduction scalars) ----------------
__global__ void init_kernel(float* __restrict__ colsum, float* __restrict__ red) {
    int t = threadIdx.x;
    colsum[t] = 0.0f;
    if (t < 8) red[t] = 0.0f;
}

// ---------------- outs GEMM: total[(l*4096+b)*256+o] = x@W[l]^T + b[l] -----
// one wave per 16x64 tile (1 A-frag reused over 4 B-frags);
// 2 layers * 256 mtiles * 4 ngroups = 2048 waves
__global__ void gemm_outs_kernel(const float* __restrict__ x,
                                 const float* __restrict__ W,
                                 const float* __restrict__ bias,
                                 float* __restrict__ total) {
    int wave = (blockIdx.x * blockDim.x + threadIdx.x) >> 5;
    int lane = threadIdx.x & 31;
    int l   = wave >> 10;            // /1024
    int rem = wave & 1023;
    int m0  = (rem >> 2) << 4;       // batch tile origin (256 mtiles)
    int n0  = (rem & 3) << 6;        // out group origin (4 groups of 64)
    int half = lane >> 4;            // 0: K pair {0,1}; 1: K pair {2,3}
    int lr   = lane & 15;

    const float* arow = x + (size_t)(m0 + lr) * IN_SZ + half * 2;
    const float* Wl   = W + (size_t)l * OUT_SZ * IN_SZ;
    const float* brow0 = Wl + (size_t)(n0 +  0 + lr) * IN_SZ + half * 2;
    const float* brow1 = Wl + (size_t)(n0 + 16 + lr) * IN_SZ + half * 2;
    const float* brow2 = Wl + (size_t)(n0 + 32 + lr) * IN_SZ + half * 2;
    const float* brow3 = Wl + (size_t)(n0 + 48 + lr) * IN_SZ + half * 2;

    v8f acc0 = {}, acc1 = {}, acc2 = {}, acc3 = {};
    for (int k = 0; k < IN_SZ; k += 4) {
        v2f a  = { arow[k],  arow[k + 1]  };
        v2f b0 = { brow0[k], brow0[k + 1] };
        v2f b1 = { brow1[k], brow1[k + 1] };
        v2f b2 = { brow2[k], brow2[k + 1] };
        v2f b3 = { brow3[k], brow3[k + 1] };
        acc0 = wmma_f32(a, b0, acc0);
        acc1 = wmma_f32(a, b1, acc1);
        acc2 = wmma_f32(a, b2, acc2);
        acc3 = wmma_f32(a, b3, acc3);
    }
    int rowbase = l * B_SZ + m0 + half * 8;
    #pragma unroll
    for (int jb = 0; jb < 4; ++jb) {
        const v8f& acc = jb == 0 ? acc0 : (jb == 1 ? acc1 : (jb == 2 ? acc2 : acc3));
        int col = n0 + jb * 16 + lr;
        float bb = bias[l * OUT_SZ + col];
        #pragma unroll
        for (int r = 0; r < 8; ++r)
            total[(size_t)(rowbase + r) * OUT_SZ + col] = acc[r] + bb;
    }
}

// ---------------- gates: softmax(x @ W_aux^T + b_aux), one wave per row ----
__global__ void gates_kernel(const float* __restrict__ x,
                             const float* __restrict__ W_aux,
                             const float* __restrict__ b_aux,
                             float* __restrict__ gates) {
    int wave = (blockIdx.x * blockDim.x + threadIdx.x) >> 5;
    int lane = threadIdx.x & 31;
    const float* xr = x + (size_t)wave * IN_SZ;
    float z0 = 0.f, z1 = 0.f;
    for (int i = lane; i < IN_SZ; i += 32) {
        float v = xr[i];
        z0 += v * W_aux[i];
        z1 += v * W_aux[IN_SZ + i];
    }
    #pragma unroll
    for (int off = 16; off; off >>= 1) {
        z0 += __shfl_down(z0, off);
        z1 += __shfl_down(z1, off);
    }
    if (lane == 0) {
        z0 += b_aux[0]; z1 += b_aux[1];
        float m  = fmaxf(z0, z1);
        float e0 = __expf(z0 - m), e1 = __expf(z1 - m);
        float inv = 1.0f / (e0 + e1);
        gates[wave * 2]     = e0 * inv;
        gates[wave * 2 + 1] = e1 * inv;
    }
}

// ---------------- combine = g0*outs0 + g1*outs1 ----------------------------
__global__ void combine_kernel(const float* __restrict__ total,
                               const float* __restrict__ gates,
                               float* __restrict__ out) {
    int b = blockIdx.x, o = threadIdx.x;
    float g0 = gates[b * 2], g1 = gates[b * 2 + 1];
    out[(size_t)b * OUT_SZ + o] = g0 * total[(size_t)b * OUT_SZ + o]
                                + g1 * total[(size_t)(B_SZ + b) * OUT_SZ + o];
}

// ---------------- per-row sum of squares + global sumsq --------------------
__global__ void rowsq_kernel(const float* __restrict__ total,
                             float* __restrict__ sq, float* __restrict__ red) {
    int wave = (blockIdx.x * blockDim.x + threadIdx.x) >> 5;
    int lane = threadIdx.x & 31;
    const float* r = total + (size_t)wave * OUT_SZ;
    float s = 0.f;
    #pragma unroll
    for (int i = lane; i < OUT_SZ; i += 32) { float v = r[i]; s += v * v; }
    #pragma unroll
    for (int off = 16; off; off >>= 1) s += __shfl_down(s, off);
    if (lane == 0) { sq[wave] = s; atomicAdd(&red[0], s); }
}

// ---------------- column sums (for ||sum_i t_i||^2) ------------------------
__global__ void colsum_kernel(const float* __restrict__ total,
                              float* __restrict__ colsum) {
    int o = threadIdx.x;
    int r0 = blockIdx.x * 128;
    float s = 0.f;
    for (int r = r0; r < r0 + 128; ++r) s += total[(size_t)r * OUT_SZ + o];
    atomicAdd(&colsum[o], s);
}

// ---------------- bandwidth + gamma_k = -1/(bw * 2^k) ----------------------
__global__ void bw_kernel(const float* __restrict__ colsum, float* __restrict__ red) {
    __shared__ float sh[256];
    int t = threadIdx.x;
    float c = colsum[t];
    sh[t] = c * c;
    __syncthreads();
    for (int off = 128; off; off >>= 1) {
        if (t < off) sh[t] += sh[t + off];
        __syncthreads();
    }
    if (t == 0) {
        float sumsq = red[0];
        // sum(dist) = 2n*sumsq - 2*||colsum||^2
        float sumd = 2.0f * (float)NTOT * sumsq - 2.0f * sh[0];
        float bw = sumd / (float)((double)NTOT * NTOT - NTOT);
        bw *= 0.25f;                       // / KERNEL_MUL^(KERNEL_NUM/2) = /4
        float b = bw;
        #pragma unroll
        for (int k = 0; k < 5; ++k) { red[2 + k] = -1.0f / b; b *= 2.0f; }
    }
}

// ---------------- pair pass: WMMA Gram + 5-bandwidth kernel sum ------------
// one wave per 64x64 tile (4 A-frags x 4 B-frags = 16 accumulators);
// only ti<=tj (symmetry), off-diagonal tiles weighted 2x
__global__ void pair_kernel(const float* __restrict__ total,
                            const float* __restrict__ sq,
                            const float* __restrict__ red) {
    int wave = (blockIdx.x * blockDim.x + threadIdx.x) >> 5;
    int lane = threadIdx.x & 31;
    int ti = wave >> 7;          // /NT64
    int tj = wave & (NT64 - 1);
    if (ti > tj) return;

    float g0 = red[2], g1 = red[3], g2 = red[4], g3 = red[5], g4 = red[6];
    int i0 = ti << 6, j0 = tj << 6;
    float sgn = ((i0 < B_SZ) == (j0 < B_SZ)) ? 1.0f : -1.0f;
    float weight = sgn * ((ti == tj) ? 1.0f : 2.0f);

    int half = lane >> 4, lr = lane & 15;
    const float* Ap[4];
    const float* Bp[4];
    #pragma unroll
    for (int t = 0; t < 4; ++t) {
        Ap[t] = total + (size_t)(i0 + t * 16 + lr) * OUT_SZ + half * 2;
        Bp[t] = total + (size_t)(j0 + t * 16 + lr) * OUT_SZ + half * 2;
    }

    v8f acc[4][4];
    #pragma unroll
    for (int ib = 0; ib < 4; ++ib)
        #pragma unroll
        for (int jb = 0; jb < 4; ++jb)
            acc[ib][jb] = (v8f){};

    for (int k = 0; k < OUT_SZ; k += 4) {
        v2f af[4], bf[4];
        #pragma unroll
        for (int t = 0; t < 4; ++t) {
            af[t] = (v2f){ Ap[t][k], Ap[t][k + 1] };
            bf[t] = (v2f){ Bp[t][k], Bp[t][k + 1] };
        }
        #pragma unroll
        for (int ib = 0; ib < 4; ++ib)
            #pragma unroll
            for (int jb = 0; jb < 4; ++jb)
                acc[ib][jb] = wmma_f32(af[ib], bf[jb], acc[ib][jb]);
    }

    // C/D layout: VGPR r -> M = r + (lane>=16 ? 8 : 0); N = lane & 15
    float sqj[4];
    #pragma unroll
    for (int jb = 0; jb < 4; ++jb) sqj[jb] = sq[j0 + jb * 16 + lr];

    float s = 0.f;
    #pragma unroll
    for (int ib = 0; ib < 4; ++ib) {
        int rbase = i0 + ib * 16 + half * 8;
        #pragma unroll
        for (int r = 0; r < 8; ++r) {
            float sqi = sq[rbase + r];
            #pragma unroll
            for (int jb = 0; jb < 4; ++jb) {
                float dist = sqi + sqj[jb] - 2.0f * acc[ib][jb][r];
                s += __expf(g0 * dist) + __expf(g1 * dist) + __expf(g2 * dist)
                   + __expf(g3 * dist) + __expf(g4 * dist);
            }
        }
    }
    s *= weight;
    #pragma unroll
    for (int off = 16; off; off >>= 1) s += __shfl_down(s, off);
    if (lane == 0) atomicAdd((float*)&red[1], s);
}

// ---------------- final scalar: -mean(XX+YY-XY-YX) -------------------------
__global__ void final_kernel(const float* __restrict__ red, float* __restrict__ out) {
    out[(size_t)B_SZ * OUT_SZ] = -red[1] / ((float)B_SZ * (float)B_SZ);
}

extern "C" void kernel_launch(void* const* d_in, const int* in_sizes, int n_in,
                              void* d_out, int out_size, void* d_ws, size_t ws_size,
                              hipStream_t stream) {
    const float* x     = (const float*)d_in[0];   // (4096, 512)
    const float* W_aux = (const float*)d_in[1];   // (2, 512)
    const float* b_aux = (const float*)d_in[2];   // (2,)
    const float* W     = (const float*)d_in[3];   // (2, 256, 512)
    const float* bias  = (const float*)d_in[4];   // (2, 256)
    float* out = (float*)d_out;                   // 4096*256 combine + 1 scalar

    float* ws     = (float*)d_ws;
    float* total  = ws;                           // 8192*256
    float* sq     = ws + (size_t)NTOT * OUT_SZ;   // 8192
    float* gates  = sq + NTOT;                    // 8192
    float* colsum = gates + NTOT;                 // 256
    float* red    = colsum + 256;                 // 8: [0]=sumsq [1]=ksum [2..6]=gamma

    init_kernel<<<1, 256, 0, stream>>>(colsum, red);
    gemm_outs_kernel<<<256, 256, 0, stream>>>(x, W, bias, total);    // 2048 waves
    gates_kernel<<<512, 256, 0, stream>>>(x, W_aux, b_aux, gates);   // 4096 waves
    combine_kernel<<<B_SZ, OUT_SZ, 0, stream>>>(total, gates, out);
    rowsq_kernel<<<1024, 256, 0, stream>>>(total, sq, red);          // 8192 waves
    colsum_kernel<<<64, 256, 0, stream>>>(total, colsum);
    bw_kernel<<<1, 256, 0, stream>>>(colsum, red);
    pair_kernel<<<2048, 256, 0, stream>>>(total, sq, red);           // 16384 waves
    final_kernel<<<1, 1, 0, stream>>>(red, out);
}